// GAT_2353642078894
// MI455X (gfx1250) — compile-verified
//
#include <hip/hip_runtime.h>
#include <math.h>

#define N_NODES 50000
#define N_EDGES 800000
#define NEG_SLOPE 0.2f

typedef float v2f __attribute__((ext_vector_type(2)));
typedef float v8f __attribute__((ext_vector_type(8)));

__device__ __forceinline__ float leaky(float x) {
    return x > 0.0f ? x : NEG_SLOPE * x;
}

// Float atomic max via CAS loop (compile-safe on all HIP versions).
__device__ __forceinline__ void atomicMaxF(float* addr, float val) {
    int* ia = (int*)addr;
    int cur = __float_as_int(*addr);
    while (__int_as_float(cur) < val) {
        int prev = atomicCAS(ia, cur, __float_as_int(val));
        if (prev == cur) break;
        cur = prev;
    }
}

__global__ void fill_kernel(float* __restrict__ p, float v, int n) {
    int i = blockIdx.x * blockDim.x + threadIdx.x;
    if (i < n) p[i] = v;
}

// C[M,NC] = A[M,K] @ B[K,NC].  M % 16 == 0, NC % 16 == 0, K % 4 == 0.
// One 16x16 output tile per wave32, K stepped by 4 with V_WMMA_F32_16X16X4_F32.
// A frag (16x4 f32, 2 VGPR): lanes 0-15 hold K={0,1}, lanes 16-31 K={2,3}, M=lane%16.
// B frag (4x16 f32, 2 VGPR): mirrored, N=lane%16.
// D (16x16 f32, 8 VGPR): VGPR r -> M = r + 8*(lane/16), N = lane%16.
template <int K, int NC>
__global__ void gemm_wmma_f32(const float* __restrict__ A,
                              const float* __restrict__ B,
                              float* __restrict__ C, int M) {
    const int lane = threadIdx.x & 31;
    const int gw   = blockIdx.x * (blockDim.x >> 5) + (threadIdx.x >> 5);
    const int tiles_n = NC / 16;
    const int total   = (M / 16) * tiles_n;
    if (gw >= total) return;
    const int tm   = (gw / tiles_n) * 16;
    const int tn   = (gw % tiles_n) * 16;
    const int half = lane >> 4;   // 0: K pair {0,1}; 1: K pair {2,3}
    const int lid  = lane & 15;

    const float* Arow = A + (size_t)(tm + lid) * K + 2 * half;
    const float* Bcol = B + (size_t)(2 * half) * NC + tn + lid;

    v8f acc = {};
#pragma unroll 4
    for (int k = 0; k < K; k += 4) {
        v2f a, b;
        a.x = Arow[k];
        a.y = Arow[k + 1];
        b.x = Bcol[(size_t)k * NC];
        b.y = Bcol[(size_t)k * NC + NC];
        acc = __builtin_amdgcn_wmma_f32_16x16x4_f32(
            /*neg_a=*/false, a, /*neg_b=*/false, b,
            /*c_mod=*/(short)0, acc, /*reuse_a=*/false, /*reuse_b=*/false);
    }
#pragma unroll
    for (int r = 0; r < 8; ++r) {
        C[(size_t)(tm + r + 8 * half) * NC + tn + lid] = acc[r];
    }
}

// el[n,h] = dot(hbuf[n,h,:], al[h,:]); er likewise. One wave per (node, head).
template <int H, int DOUT>
__global__ void node_dots(const float* __restrict__ hbuf,
                          const float* __restrict__ al,
                          const float* __restrict__ ar,
                          float* __restrict__ el, float* __restrict__ er) {
    const int lane = threadIdx.x & 31;
    const int gw   = blockIdx.x * (blockDim.x >> 5) + (threadIdx.x >> 5);
    if (gw >= N_NODES * H) return;
    const int n = gw / H, hh = gw % H;
    const float* hp  = hbuf + (size_t)n * (H * DOUT) + hh * DOUT;
    const float* alp = al + hh * DOUT;
    const float* arp = ar + hh * DOUT;
    float sl = 0.0f, sr = 0.0f;
#pragma unroll
    for (int i = lane; i < DOUT; i += 32) {
        float v = hp[i];
        sl += v * alp[i];
        sr += v * arp[i];
    }
#pragma unroll
    for (int m = 16; m > 0; m >>= 1) {
        sl += __shfl_xor(sl, m, 32);
        sr += __shfl_xor(sr, m, 32);
    }
    if (lane == 0) { el[gw] = sl; er[gw] = sr; }
}

// Pass 1: per-edge attention logit, segment max into m[dst,h].
template <int H>
__global__ void edge_max(const float* __restrict__ el,
                         const float* __restrict__ er,
                         const int* __restrict__ src,
                         const int* __restrict__ dst,
                         float* __restrict__ m) {
    int i = blockIdx.x * blockDim.x + threadIdx.x;
    if (i >= N_EDGES) return;
    const int sN = src[i], dN = dst[i];
#pragma unroll
    for (int hh = 0; hh < H; ++hh) {
        float e = leaky(el[sN * H + hh] + er[dN * H + hh]);
        atomicMaxF(&m[dN * H + hh], e);
    }
}

// m: -inf (no incoming edges) -> 0, per reference; also zero-init s.
__global__ void fix_m_init_s(float* __restrict__ m, float* __restrict__ s, int n) {
    int i = blockIdx.x * blockDim.x + threadIdx.x;
    if (i < n) {
        float v = m[i];
        m[i] = __builtin_isinf(v) ? 0.0f : v;
        s[i] = 0.0f;
    }
}

// Pass 2: ex = exp(e - m[dst]); segment sum into s[dst,h].
template <int H>
__global__ void edge_expsum(const float* __restrict__ el,
                            const float* __restrict__ er,
                            const float* __restrict__ m,
                            const int* __restrict__ src,
                            const int* __restrict__ dst,
                            float* __restrict__ s) {
    int i = blockIdx.x * blockDim.x + threadIdx.x;
    if (i >= N_EDGES) return;
    const int sN = src[i], dN = dst[i];
#pragma unroll
    for (int hh = 0; hh < H; ++hh) {
        float e = leaky(el[sN * H + hh] + er[dN * H + hh]);
        atomicAdd(&s[dN * H + hh], __expf(e - m[dN * H + hh]));
    }
}

// Pass 3: agg[dst] += alpha * h[src].  One wave per edge, 8 elems/lane.
template <int H, int DOUT>
__global__ void edge_scatter(const float* __restrict__ hbuf,
                             const float* __restrict__ el,
                             const float* __restrict__ er,
                             const float* __restrict__ m,
                             const float* __restrict__ s,
                             const int* __restrict__ src,
                             const int* __restrict__ dst,
                             float* __restrict__ agg) {
    const int lane = threadIdx.x & 31;
    const int gw   = blockIdx.x * (blockDim.x >> 5) + (threadIdx.x >> 5);
    if (gw >= N_EDGES) return;
    const int sN = src[gw], dN = dst[gw];
    float alpha[H];
#pragma unroll
    for (int hh = 0; hh < H; ++hh) {
        float e = leaky(el[sN * H + hh] + er[dN * H + hh]);
        alpha[hh] = __expf(e - m[dN * H + hh]) / s[dN * H + hh];
    }
    constexpr int NC = H * DOUT;
    const float* hs = hbuf + (size_t)sN * NC;
    float* ad = agg + (size_t)dN * NC;
#pragma unroll
    for (int i = 0; i < NC / 32; ++i) {
        int j = i * 32 + lane;
        atomicAdd(&ad[j], hs[j] * alpha[j / DOUT]);
    }
}

// x = relu(agg + b)   (in-place OK; NC == 256)
__global__ void bias_relu_kernel(const float* __restrict__ agg,
                                 const float* __restrict__ b,
                                 float* __restrict__ out, int n) {
    int i = blockIdx.x * blockDim.x + threadIdx.x;
    if (i < n) {
        float v = agg[i] + b[i & 255];
        out[i] = v > 0.0f ? v : 0.0f;
    }
}

// out += b1; row 0 -> 0.  (NC == 256)
__global__ void bias_zero0_kernel(float* __restrict__ out,
                                  const float* __restrict__ b, int n) {
    int i = blockIdx.x * blockDim.x + threadIdx.x;
    if (i < n) {
        float v = out[i] + b[i & 255];
        out[i] = (i < 256) ? 0.0f : v;
    }
}

extern "C" void kernel_launch(void* const* d_in, const int* in_sizes, int n_in,
                              void* d_out, int out_size, void* d_ws, size_t ws_size,
                              hipStream_t stream) {
    const float* emb = (const float*)d_in[0];
    const float* W0  = (const float*)d_in[1];
    const float* al0 = (const float*)d_in[2];
    const float* ar0 = (const float*)d_in[3];
    const float* b0  = (const float*)d_in[4];
    const float* W1  = (const float*)d_in[5];
    const float* al1 = (const float*)d_in[6];
    const float* ar1 = (const float*)d_in[7];
    const float* b1  = (const float*)d_in[8];
    const int*   src = (const int*)d_in[9];
    const int*   dst = (const int*)d_in[10];
    float* out = (float*)d_out;

    float* ws   = (float*)d_ws;
    float* hbuf = ws;                                   // N*256 (h0, later h1)
    float* agg  = hbuf + (size_t)N_NODES * 256;         // N*256 (agg0 -> x1 in place)
    float* el   = agg + (size_t)N_NODES * 256;          // N*2
    float* er   = el + N_NODES * 2;                     // N*2
    float* m    = er + N_NODES * 2;                     // N*2
    float* s    = m + N_NODES * 2;                      // N*2

    const int TPB = 256;
    const int WPB = TPB / 32;  // waves per block
    const int nEdgeBlk  = (N_EDGES + TPB - 1) / TPB;          // 3125
    const int nEdgeWblk = (N_EDGES + WPB - 1) / WPB;          // 100000
    const int nElem     = N_NODES * 256;
    const int nElemBlk  = (nElem + TPB - 1) / TPB;

    // ================= Layer 0 (H=2, Dout=128, K=128) =================
    {
        const int tiles = (N_NODES / 16) * (256 / 16);        // 50000
        gemm_wmma_f32<128, 256><<<(tiles + WPB - 1) / WPB, TPB, 0, stream>>>(
            emb, W0, hbuf, N_NODES);
    }
    node_dots<2, 128><<<(N_NODES * 2 + WPB - 1) / WPB, TPB, 0, stream>>>(
        hbuf, al0, ar0, el, er);
    fill_kernel<<<(N_NODES * 2 + TPB - 1) / TPB, TPB, 0, stream>>>(
        m, -INFINITY, N_NODES * 2);
    fill_kernel<<<nElemBlk, TPB, 0, stream>>>(agg, 0.0f, nElem);
    edge_max<2><<<nEdgeBlk, TPB, 0, stream>>>(el, er, src, dst, m);
    fix_m_init_s<<<(N_NODES * 2 + TPB - 1) / TPB, TPB, 0, stream>>>(
        m, s, N_NODES * 2);
    edge_expsum<2><<<nEdgeBlk, TPB, 0, stream>>>(el, er, m, src, dst, s);
    edge_scatter<2, 128><<<nEdgeWblk, TPB, 0, stream>>>(
        hbuf, el, er, m, s, src, dst, agg);
    bias_relu_kernel<<<nElemBlk, TPB, 0, stream>>>(agg, b0, agg, nElem);  // x1 in agg

    // ================= Layer 1 (H=1, Dout=256, K=256) =================
    {
        const int tiles = (N_NODES / 16) * (256 / 16);
        gemm_wmma_f32<256, 256><<<(tiles + WPB - 1) / WPB, TPB, 0, stream>>>(
            agg, W1, hbuf, N_NODES);
    }
    node_dots<1, 256><<<(N_NODES + WPB - 1) / WPB, TPB, 0, stream>>>(
        hbuf, al1, ar1, el, er);
    fill_kernel<<<(N_NODES + TPB - 1) / TPB, TPB, 0, stream>>>(
        m, -INFINITY, N_NODES);
    fill_kernel<<<nElemBlk, TPB, 0, stream>>>(out, 0.0f, nElem);
    edge_max<1><<<nEdgeBlk, TPB, 0, stream>>>(el, er, src, dst, m);
    fix_m_init_s<<<(N_NODES + TPB - 1) / TPB, TPB, 0, stream>>>(m, s, N_NODES);
    edge_expsum<1><<<nEdgeBlk, TPB, 0, stream>>>(el, er, m, src, dst, s);
    edge_scatter<1, 256><<<nEdgeWblk, TPB, 0, stream>>>(
        hbuf, el, er, m, s, src, dst, out);
    bias_zero0_kernel<<<nElemBlk, TPB, 0, stream>>>(out, b1, nElem);
}